// VectorQuantizer_68719477175
// MI455X (gfx1250) — compile-verified
//
#include <hip/hip_runtime.h>
#include <hip/hip_bf16.h>
#include <math.h>

typedef float v2f __attribute__((ext_vector_type(2)));
typedef float v8f __attribute__((ext_vector_type(8)));

#define N_PTS   8192      // 8*32*32 spatial positions
#define N_CODE  8192      // codebook entries
#define E_DIM   256
#define PITCH   260       // LDS row pitch (floats): 260 % 64 == 4 -> conflict-free

// ---------------- reductions ----------------
__device__ __forceinline__ float warpSum(float v) {
#pragma unroll
  for (int o = 16; o; o >>= 1) v += __shfl_xor(v, o, 32);
  return v;
}

__device__ float blockSum(float v) {
  __shared__ float red[8];
  const int lane = threadIdx.x & 31, w = threadIdx.x >> 5;
  v = warpSum(v);
  if (lane == 0) red[w] = v;
  __syncthreads();
  float r = (threadIdx.x < 8) ? red[threadIdx.x] : 0.f;
  if (w == 0) {
#pragma unroll
    for (int o = 4; o; o >>= 1) r += __shfl_xor(r, o, 32);
    if (lane == 0) red[0] = r;
  }
  __syncthreads();
  float out = red[0];
  __syncthreads();
  return out;
}

// ---------------- init scratch ----------------
__global__ void init_kernel(float* colsum, float* vq_sum) {
  int i = blockIdx.x * blockDim.x + threadIdx.x;
  if (i < N_CODE) colsum[i] = 0.f;
  if (i == 0) *vq_sum = 0.f;
}

// ---------------- L2-normalize z (NCHW -> row-major [N, 256]) ----------------
__global__ __launch_bounds__(256) void norm_z_kernel(const float* __restrict__ z,
                                                     float* __restrict__ Zn,
                                                     float* __restrict__ zz) {
  const int n = blockIdx.x, c = threadIdx.x;
  const int b = n >> 10, rem = n & 1023;           // rem = h*32+w
  const float v = z[b * 262144 + c * 1024 + rem];
  const float ss = blockSum(v * v);
  const float inv = 1.f / fmaxf(sqrtf(ss), 1e-12f);
  Zn[n * E_DIM + c] = v * inv;
  if (c == 0) zz[n] = ss * inv * inv;              // ||z_norm||^2 (≈1, exact per ref)
}

// ---------------- L2-normalize embedding ----------------
__global__ __launch_bounds__(256) void norm_e_kernel(const float* __restrict__ emb,
                                                     float* __restrict__ En,
                                                     float* __restrict__ ee) {
  const int n = blockIdx.x, c = threadIdx.x;
  const float v = emb[n * E_DIM + c];
  const float ss = blockSum(v * v);
  const float inv = 1.f / fmaxf(sqrtf(ss), 1e-12f);
  En[n * E_DIM + c] = v * inv;
  if (c == 0) ee[n] = ss * inv * inv;
}

// ---------------- pass 1: per-row argmax(f) + online softmax stats ----------------
// f_nk = -d_nk / T = 200*dot - 100*(zz_n + ee_k). Block = 16 z-rows, 8 waves split col tiles.
__global__ __launch_bounds__(256) void vq_rowstats_kernel(
    const float* __restrict__ Zn, const float* __restrict__ En,
    const float* __restrict__ zz, const float* __restrict__ ee,
    float* __restrict__ lse, float* __restrict__ ent, int* __restrict__ idx) {
  __shared__ float smem[16 * PITCH];
  const int R = blockIdx.x * 16;
  const int tid = threadIdx.x;

  for (int i = tid; i < 16 * E_DIM; i += 256) {
    int r = i >> 8, col = i & 255;
    smem[r * PITCH + col] = Zn[(R + r) * E_DIM + col];
  }
  __syncthreads();

  const int w = tid >> 5, lane = tid & 31;
  const int hi = lane >> 4, lo = lane & 15;

  float rm[8], rs[8], rw[8], bf[8], zzv[8];
  int bi[8];
#pragma unroll
  for (int j = 0; j < 8; j++) {
    rm[j] = -INFINITY; rs[j] = 0.f; rw[j] = 0.f; bf[j] = -INFINITY; bi[j] = 0x7fffffff;
    zzv[j] = zz[R + j + 8 * hi];                   // C-frag row = j + 8*(lane>=16)
  }

  const float* Ap = smem + lo * PITCH + 2 * hi;    // A-frag: m = lane&15, k = 2*(lane>=16)+{0,1}
  for (int tile = w; tile < N_CODE / 16; tile += 8) {
    const int nlane = tile * 16 + lo;              // B/C-frag col
    const float een = ee[nlane];
    const float* Bp = En + nlane * E_DIM + 2 * hi;
    v8f c = {};
#pragma unroll 8
    for (int k0 = 0; k0 < E_DIM; k0 += 4) {
      v2f a = *(const v2f*)(Ap + k0);
      v2f b = *(const v2f*)(Bp + k0);
      c = __builtin_amdgcn_wmma_f32_16x16x4_f32(false, a, false, b, (short)0, c, false, false);
    }
#pragma unroll
    for (int j = 0; j < 8; j++) {
      const float x = 200.f * c[j] - 100.f * (zzv[j] + een);
      if (x > bf[j]) { bf[j] = x; bi[j] = nlane; } // first occurrence wins (n increases)
      const float nm = fmaxf(rm[j], x);
      const float e1 = __expf(rm[j] - nm), e2 = __expf(x - nm);
      rs[j] = rs[j] * e1 + e2;
      rw[j] = rw[j] * e1 + e2 * x;
      rm[j] = nm;
    }
  }

  // merge across the 16 lanes of each half (cols interleave mod 16)
#pragma unroll
  for (int off = 1; off < 16; off <<= 1) {
#pragma unroll
    for (int j = 0; j < 8; j++) {
      float om = __shfl_xor(rm[j], off, 32);
      float os = __shfl_xor(rs[j], off, 32);
      float ow = __shfl_xor(rw[j], off, 32);
      float of = __shfl_xor(bf[j], off, 32);
      int   oi = __shfl_xor(bi[j], off, 32);
      float nm = fmaxf(rm[j], om);
      float e1 = __expf(rm[j] - nm), e2 = __expf(om - nm);
      rs[j] = rs[j] * e1 + os * e2;
      rw[j] = rw[j] * e1 + ow * e2;
      rm[j] = nm;
      if (of > bf[j] || (of == bf[j] && oi < bi[j])) { bf[j] = of; bi[j] = oi; }
    }
  }
  __syncthreads();  // done reading As; reuse smem for cross-wave stats

  float* sm = smem;          // [16 rows][8 waves]
  float* sS = smem + 128;
  float* sW = smem + 256;
  float* sF = smem + 384;
  int*   sI = (int*)(smem + 512);
  if (lo == 0) {
    const int rbase = 8 * hi;
#pragma unroll
    for (int j = 0; j < 8; j++) {
      const int r = rbase + j;
      sm[r * 8 + w] = rm[j]; sS[r * 8 + w] = rs[j]; sW[r * 8 + w] = rw[j];
      sF[r * 8 + w] = bf[j]; sI[r * 8 + w] = bi[j];
    }
  }
  __syncthreads();
  if (tid < 16) {
    const int r = tid;
    float m = -INFINITY, S = 0.f, W = 0.f, f = -INFINITY;
    int I = 0x7fffffff;
#pragma unroll
    for (int q = 0; q < 8; q++) {
      float om = sm[r * 8 + q], os = sS[r * 8 + q], ow = sW[r * 8 + q], of = sF[r * 8 + q];
      int oi = sI[r * 8 + q];
      float nm = fmaxf(m, om);
      float e1 = __expf(m - nm), e2 = __expf(om - nm);
      S = S * e1 + os * e2; W = W * e1 + ow * e2; m = nm;
      if (of > f || (of == f && oi < I)) { f = of; I = oi; }
    }
    const float L = m + __logf(S);                 // logsumexp_n
    lse[R + r] = L;
    ent[R + r] = L - W / S;                        // -sum_k p*logp (shift-invariant)
    idx[R + r] = I;
  }
}

// ---------------- pass 2: column sums of probs (avg_probs numerator) ----------------
__global__ __launch_bounds__(256) void vq_colsum_kernel(
    const float* __restrict__ Zn, const float* __restrict__ En,
    const float* __restrict__ zz, const float* __restrict__ ee,
    const float* __restrict__ lse, float* __restrict__ colsum) {
  __shared__ float Es[16 * PITCH];
  const int Cn = blockIdx.x * 16;                  // col tile base
  const int rt0 = blockIdx.y * 64;                 // 64 row tiles = 1024 rows per block
  const int tid = threadIdx.x;

  for (int i = tid; i < 16 * E_DIM; i += 256) {
    int r = i >> 8, col = i & 255;
    Es[r * PITCH + col] = En[(Cn + r) * E_DIM + col];
  }
  __syncthreads();

  const int w = tid >> 5, lane = tid & 31;
  const int hi = lane >> 4, lo = lane & 15;
  const int n = Cn + lo;                           // this lane's output column
  const float een = ee[n];
  const float* Bp = Es + lo * PITCH + 2 * hi;

  float acc = 0.f;
  for (int t = w; t < 64; t += 8) {
    const int rbase = (rt0 + t) * 16;
    const float* Ap = Zn + (rbase + lo) * E_DIM + 2 * hi;
    v8f c = {};
#pragma unroll 8
    for (int k0 = 0; k0 < E_DIM; k0 += 4) {
      v2f a = *(const v2f*)(Ap + k0);
      v2f b = *(const v2f*)(Bp + k0);
      c = __builtin_amdgcn_wmma_f32_16x16x4_f32(false, a, false, b, (short)0, c, false, false);
    }
#pragma unroll
    for (int j = 0; j < 8; j++) {
      const int m = rbase + j + 8 * hi;
      const float x = 200.f * c[j] - 100.f * (zz[m] + een);
      acc += __expf(x - lse[m]);
    }
  }
  acc += __shfl_xor(acc, 16, 32);                  // combine row halves (same column)
  if (hi == 0) atomicAdd(&colsum[n], acc);
}

// ---------------- scatter z_q (NHWC rows -> NCHW out) + vq loss + idx out ----------------
__global__ __launch_bounds__(256) void vq_output_kernel(
    const float* __restrict__ Zn, const float* __restrict__ En,
    const int* __restrict__ idx, float* __restrict__ out_zq,
    float* __restrict__ out_idx, float* __restrict__ vq_sum) {
  const int n = blockIdx.x, c = threadIdx.x;
  const int k = idx[n];
  const float e = En[k * E_DIM + c];
  const float d = e - Zn[n * E_DIM + c];
  const float s = blockSum(d * d);
  const int b = n >> 10, rem = n & 1023;
  out_zq[b * 262144 + c * 1024 + rem] = e;         // z_q_st == z_q numerically (STE)
  if (c == 0) {
    atomicAdd(vq_sum, s);
    out_idx[n] = (float)k;
  }
}

// ---------------- finalize scalars ----------------
__global__ __launch_bounds__(256) void finalize_kernel(
    const float* __restrict__ ent, const float* __restrict__ colsum,
    const float* __restrict__ vq_sum, float* __restrict__ out_scalars) {
  const int tid = threadIdx.x;
  float se = 0.f, ae = 0.f;
  for (int k = tid; k < N_CODE; k += 256) {
    se += ent[k];
    const float q = colsum[k] * (1.f / 8192.f);    // avg_probs[k]
    ae += q * __logf(q + 1e-5f);
  }
  se = blockSum(se);
  ae = blockSum(ae);
  if (tid == 0) {
    const float sample_entropy = se * (1.f / 8192.f);
    const float avg_entropy = -ae;
    const float vq = vq_sum[0] * (1.f / (8192.f * 256.f));
    out_scalars[0] = vq;                           // vq_loss
    out_scalars[1] = 0.25f * vq;                   // commit_loss = BETA * same value
    out_scalars[2] = 0.1f * (sample_entropy - avg_entropy); // entropy_loss
  }
}

extern "C" void kernel_launch(void* const* d_in, const int* in_sizes, int n_in,
                              void* d_out, int out_size, void* d_ws, size_t ws_size,
                              hipStream_t stream) {
  const float* z   = (const float*)d_in[0];   // [8,256,32,32]
  const float* emb = (const float*)d_in[1];   // [8192,256]

  float* ws = (float*)d_ws;
  float* Zn     = ws;                         // 2,097,152
  float* En     = ws + 2097152;               // 2,097,152
  float* zz     = ws + 4194304;               // 8192
  float* ee     = ws + 4202496;               // 8192
  float* lse    = ws + 4210688;               // 8192
  float* ent    = ws + 4218880;               // 8192
  int*   idx    = (int*)(ws + 4227072);       // 8192
  float* colsum = ws + 4235264;               // 8192
  float* vq_sum = ws + 4243456;               // 1

  float* out         = (float*)d_out;
  float* out_zq      = out;                   // 2,097,152
  float* out_scalars = out + 2097152;         // vq, commit, entropy
  float* out_idx     = out + 2097155;         // 8192 (indices as float)

  init_kernel<<<32, 256, 0, stream>>>(colsum, vq_sum);
  norm_z_kernel<<<N_PTS, 256, 0, stream>>>(z, Zn, zz);
  norm_e_kernel<<<N_CODE, 256, 0, stream>>>(emb, En, ee);
  vq_rowstats_kernel<<<N_PTS / 16, 256, 0, stream>>>(Zn, En, zz, ee, lse, ent, idx);
  vq_colsum_kernel<<<dim3(N_CODE / 16, 8), 256, 0, stream>>>(Zn, En, zz, ee, lse, colsum);
  vq_output_kernel<<<N_PTS, 256, 0, stream>>>(Zn, En, idx, out_zq, out_idx, vq_sum);
  finalize_kernel<<<1, 256, 0, stream>>>(ent, colsum, vq_sum, out_scalars);
}